// TPSWarper_58626303590806
// MI455X (gfx1250) — compile-verified
//
#include <hip/hip_runtime.h>
#include <math.h>

typedef __attribute__((ext_vector_type(2))) float v2f;
typedef __attribute__((ext_vector_type(8))) float v8f;

#define W_ 512
#define H_ 512
#define HW_ (W_ * H_)
#define NCTRL 16
#define NP 19          // N + 3
#define NAUG 21        // 19 matrix cols + 2 rhs cols
#define PSTRIDE 40     // floats per batch in workspace (19*2 params, padded)

// ---------------------------------------------------------------------------
// Kernel 1: per-batch TPS solve.  L (19x19) params = rhs (19x2), Gauss-Jordan
// with partial pivoting in double precision in LDS.  One wave per batch.
// control points = target landmarks, rhs = source landmarks (per reference).
// ---------------------------------------------------------------------------
__global__ void tps_solve_kernel(const float* __restrict__ src,
                                 const float* __restrict__ tgt,
                                 float* __restrict__ params) {
    const int b   = blockIdx.x;
    const int tid = threadIdx.x;

    __shared__ double A[NP][NAUG];
    __shared__ int piv;

    const float* t = tgt + b * NCTRL * 2;
    const float* s = src + b * NCTRL * 2;

    // Build augmented system
    for (int idx = tid; idx < NP * NAUG; idx += 32) {
        const int i = idx / NAUG;
        const int j = idx % NAUG;
        double v = 0.0;
        if (j < NP) {
            if (i < NCTRL && j < NCTRL) {
                double dx = (double)t[i * 2 + 0] - (double)t[j * 2 + 0];
                double dy = (double)t[i * 2 + 1] - (double)t[j * 2 + 1];
                double sq = dx * dx + dy * dy;
                double m  = sq > 1e-12 ? sq : 1e-12;
                v = 0.5 * m * log(m);          // d^2 * log(d), d = sqrt(m)
                if (i == j) v += 1e-4;         // ridge
            } else if (i < NCTRL) {            // P block
                v = (j == NCTRL) ? 1.0
                    : (j == NCTRL + 1 ? (double)t[i * 2 + 0]
                                      : (double)t[i * 2 + 1]);
            } else if (j < NCTRL) {            // P^T block
                v = (i == NCTRL) ? 1.0
                    : (i == NCTRL + 1 ? (double)t[j * 2 + 0]
                                      : (double)t[j * 2 + 1]);
            } else {
                v = (i == j) ? 1e-4 : 0.0;     // ridge on zero block diag
            }
        } else {                               // rhs columns
            const int c = j - NP;
            v = (i < NCTRL) ? (double)s[i * 2 + c] : 0.0;
        }
        A[i][j] = v;
    }
    __syncthreads();

    // Gauss-Jordan with partial pivoting
    for (int col = 0; col < NP; ++col) {
        if (tid == 0) {
            int p = col;
            double best = fabs(A[col][col]);
            for (int r = col + 1; r < NP; ++r) {
                double av = fabs(A[r][col]);
                if (av > best) { best = av; p = r; }
            }
            piv = p;
        }
        __syncthreads();
        if (tid < NAUG) {                      // swap rows col <-> piv
            double tmp   = A[col][tid];
            A[col][tid]  = A[piv][tid];
            A[piv][tid]  = tmp;
        }
        __syncthreads();
        if (tid < NP && tid != col) {
            double f = A[tid][col] / A[col][col];
            for (int j = col; j < NAUG; ++j)
                A[tid][j] -= f * A[col][j];
        }
        __syncthreads();
    }

    if (tid < NP) {
        double d = A[tid][tid];
        params[b * PSTRIDE + tid * 2 + 0] = (float)(A[tid][NP + 0] / d);
        params[b * PSTRIDE + tid * 2 + 1] = (float)(A[tid][NP + 1] / d);
    }
}

// ---------------------------------------------------------------------------
// Kernel 2: TPS evaluation via WMMA + bilinear sample.
// Each wave = 32 pixels = two 16-pixel tiles.  Contraction K(16x16)·W(16x2)
// done as 4 chained v_wmma_f32_16x16x4_f32 per tile:
//   A (16x4 f32): rows 0/1 = w_x/w_y over the 4-control chunk
//                 (lane=M; VGPR0=K0, VGPR1=K1; lanes>=16 carry K2/K3)
//   B (4x16 f32): rbf(pixel, control); VGPR0 lanes0-15 = K-row0,
//                 lanes16-31 = K-row2; VGPR1 = rows 1/3; N = pixel = lane%16
//   D: acc[0][laneN] = coord_x(pixel), acc[1][laneN] = coord_y(pixel)
// ---------------------------------------------------------------------------
__device__ __forceinline__ float rbf_f(float gx, float gy, float cx, float cy) {
    float dx = gx - cx;
    float dy = gy - cy;
    float sq = fmaf(dx, dx, dy * dy);
    float m  = fmaxf(sq, 1e-12f);
    return 0.5f * m * __logf(m);               // d^2 * log(d)
}

__global__ void __launch_bounds__(256)
tps_warp_kernel(const float* __restrict__ cloth,
                const float* __restrict__ tgt,
                const float* __restrict__ params,
                float* __restrict__ out) {
    const int b        = blockIdx.y;
    const int lane     = threadIdx.x & 31;
    const int waveBase = blockIdx.x * 256 + (threadIdx.x >> 5) * 32;
    const int m        = lane & 15;
    const int half     = lane >> 4;

    const float* ctl = tgt + b * NCTRL * 2;
    const float* prm = params + b * PSTRIDE;

    const float step = 2.0f / 511.0f;
    const int p0 = waveBase + m;           // tile0 pixel this lane feeds B for
    const int p1 = waveBase + 16 + m;      // tile1 pixel
    const float gx0 = (float)(p0 & (W_ - 1)) * step - 1.0f;
    const float gy0 = (float)(p0 >> 9)       * step - 1.0f;
    const float gx1 = (float)(p1 & (W_ - 1)) * step - 1.0f;
    const float gy1 = (float)(p1 >> 9)       * step - 1.0f;

    v8f acc0 = {0.f, 0.f, 0.f, 0.f, 0.f, 0.f, 0.f, 0.f};
    v8f acc1 = {0.f, 0.f, 0.f, 0.f, 0.f, 0.f, 0.f, 0.f};

#pragma unroll
    for (int k0 = 0; k0 < NCTRL; k0 += 4) {
        const int kv0 = k0 + (half ? 2 : 0);   // control feeding VGPR0 of A/B
        const int kv1 = k0 + (half ? 3 : 1);   // control feeding VGPR1 of A/B

        const float c0x = ctl[kv0 * 2 + 0], c0y = ctl[kv0 * 2 + 1];
        const float c1x = ctl[kv1 * 2 + 0], c1y = ctl[kv1 * 2 + 1];

        v2f a;
        {
            const float w0x = prm[kv0 * 2 + 0], w0y = prm[kv0 * 2 + 1];
            const float w1x = prm[kv1 * 2 + 0], w1y = prm[kv1 * 2 + 1];
            a.x = (m == 0) ? w0x : ((m == 1) ? w0y : 0.0f);
            a.y = (m == 0) ? w1x : ((m == 1) ? w1y : 0.0f);
        }

        v2f b0, b1;
        b0.x = rbf_f(gx0, gy0, c0x, c0y);
        b0.y = rbf_f(gx0, gy0, c1x, c1y);
        b1.x = rbf_f(gx1, gy1, c0x, c0y);
        b1.y = rbf_f(gx1, gy1, c1x, c1y);

        acc0 = __builtin_amdgcn_wmma_f32_16x16x4_f32(
            false, a, false, b0, (short)0, acc0, false, false);
        acc1 = __builtin_amdgcn_wmma_f32_16x16x4_f32(
            false, a, false, b1, (short)0, acc1, false, false);
    }

    // gather per-pixel coords: lane l<16 -> tile0 pixel l, lane l>=16 ->
    // tile1 pixel l-16 (value lives in lane l-16 of acc1)
    const float sx0 = __shfl(acc0[0], m, 32);
    const float sy0 = __shfl(acc0[1], m, 32);
    const float sx1 = __shfl(acc1[0], m, 32);
    const float sy1 = __shfl(acc1[1], m, 32);

    const float gxm = half ? gx1 : gx0;
    const float gym = half ? gy1 : gy0;
    float cx = (half ? sx1 : sx0) + prm[32] + prm[34] * gxm + prm[36] * gym;
    float cy = (half ? sy1 : sy0) + prm[33] + prm[35] * gxm + prm[37] * gym;

    // bilinear sample, align_corners=True, zeros padding
    const float ix  = (cx + 1.0f) * 0.5f * (float)(W_ - 1);
    const float iy  = (cy + 1.0f) * 0.5f * (float)(H_ - 1);
    const float x0f = floorf(ix);
    const float y0f = floorf(iy);
    const float wx1 = ix - x0f, wy1 = iy - y0f;
    const float wx0 = 1.0f - wx1, wy0 = 1.0f - wy1;

    const size_t imgBase = (size_t)b * 3 * HW_;
    const float* img = cloth + imgBase;
    const int myPix = waveBase + lane;

    float o0 = 0.0f, o1 = 0.0f, o2 = 0.0f;
#pragma unroll
    for (int dy = 0; dy < 2; ++dy) {
        const float yf  = y0f + (float)dy;
        const float wy  = dy ? wy1 : wy0;
        const bool  vy  = (yf >= 0.0f) && (yf <= (float)(H_ - 1));
        const float yfc = fminf(fmaxf(yf, 0.0f), (float)(H_ - 1));
        const int   yi  = (int)yfc;
#pragma unroll
        for (int dx = 0; dx < 2; ++dx) {
            const float xf  = x0f + (float)dx;
            const float wx  = dx ? wx1 : wx0;
            const bool  vx  = (xf >= 0.0f) && (xf <= (float)(W_ - 1));
            const float xfc = fminf(fmaxf(xf, 0.0f), (float)(W_ - 1));
            const int   xi  = (int)xfc;
            const float wgt = wy * wx * ((vy && vx) ? 1.0f : 0.0f);
            const int   off = yi * W_ + xi;
            o0 += img[off] * wgt;
            o1 += img[HW_ + off] * wgt;
            o2 += img[2 * HW_ + off] * wgt;
        }
    }
    out[imgBase + myPix]           = o0;
    out[imgBase + HW_ + myPix]     = o1;
    out[imgBase + 2 * HW_ + myPix] = o2;
}

// ---------------------------------------------------------------------------
extern "C" void kernel_launch(void* const* d_in, const int* in_sizes, int n_in,
                              void* d_out, int out_size, void* d_ws, size_t ws_size,
                              hipStream_t stream) {
    const float* cloth = (const float*)d_in[0];
    const float* src   = (const float*)d_in[1];
    const float* tgt   = (const float*)d_in[2];
    float* out    = (float*)d_out;
    float* params = (float*)d_ws;          // B * PSTRIDE floats (16*40 = 2.5 KB)

    const int B = in_sizes[1] / (NCTRL * 2);   // 16

    tps_solve_kernel<<<B, 32, 0, stream>>>(src, tgt, params);

    dim3 grid(HW_ / 256, B);
    tps_warp_kernel<<<grid, 256, 0, stream>>>(cloth, tgt, params, out);
}